// OverAll_16922171146745
// MI455X (gfx1250) — compile-verified
//
#include <hip/hip_runtime.h>
#include <math.h>

#define Nn     100000
#define Ee     400000
#define NNZn   800000
#define ADJn   400000
#define Dd     100
#define DOUTd  300
#define NPAD   320      // padded N dimension for gate_k packing
#define NPROXY 64

typedef __attribute__((ext_vector_type(2))) float v2f;
typedef __attribute__((ext_vector_type(8))) float v8f;

static inline int cdiv(long a, long b) { return (int)((a + b - 1) / b); }

// ---------------------------------------------------------------- utilities

__device__ inline void atomicMaxF(float* addr, float val) {
    unsigned int* u = (unsigned int*)addr;
    unsigned int old = *u;
    while (__uint_as_float(old) < val) {
        unsigned int assumed = old;
        old = atomicCAS(u, assumed, __float_as_uint(val));
        if (old == assumed) break;
    }
}

__global__ void k_fill(float* p, float v, int n) {
    int i = blockIdx.x * blockDim.x + threadIdx.x;
    if (i < n) p[i] = v;
}

// ------------------------------------------- sparse row mean (srsm) scatter

__global__ void k_count(const int* __restrict__ idx, float* __restrict__ cnt, int nnz) {
    int i = blockIdx.x * blockDim.x + threadIdx.x;
    if (i < nnz) atomicAdd(&cnt[idx[2 * i]], 1.0f);
}

__global__ void k_srsm_scatter(const int* __restrict__ idx, const float* __restrict__ emb,
                               const float* __restrict__ cnt, float* __restrict__ outf, int nnz) {
    long t = (long)blockIdx.x * blockDim.x + threadIdx.x;
    if (t >= (long)nnz * 25) return;
    int i = (int)(t / 25), d0 = (int)(t % 25) * 4;
    int row = idx[2 * i], col = idx[2 * i + 1];
    float inv = 1.0f / cnt[row];
    float4 e4 = *(const float4*)(emb + (long)col * Dd + d0);
    float* o = outf + (long)row * Dd + d0;
    atomicAdd(o + 0, e4.x * inv); atomicAdd(o + 1, e4.y * inv);
    atomicAdd(o + 2, e4.z * inv); atomicAdd(o + 3, e4.w * inv);
}

// -------------------------------------------------- per-edge relation sums

__global__ void k_rels_scatter(const int* __restrict__ idx, const float* __restrict__ val,
                               const float* __restrict__ emb, float* __restrict__ rels, int nnz) {
    long t = (long)blockIdx.x * blockDim.x + threadIdx.x;
    if (t >= (long)nnz * 25) return;
    int i = (int)(t / 25), d0 = (int)(t % 25) * 4;
    int row = idx[2 * i], col = idx[2 * i + 1];
    float v = val[i];
    float4 e4 = *(const float4*)(emb + (long)col * Dd + d0);
    float* o = rels + (long)row * Dd + d0;
    atomicAdd(o + 0, e4.x * v); atomicAdd(o + 1, e4.y * v);
    atomicAdd(o + 2, e4.z * v); atomicAdd(o + 3, e4.w * v);
}

// wave(32)-per-row L2 normalize: out[r] = in[r] / max(||in[r]||, 1e-12)
__global__ void k_l2norm_rows(const float* __restrict__ in, float* __restrict__ out,
                              int rows, int cols) {
    int wid = threadIdx.x >> 5, lane = threadIdx.x & 31;
    int row = blockIdx.x * (blockDim.x >> 5) + wid;
    if (row >= rows) return;
    const float* rp = in + (long)row * cols;
    float s = 0.0f;
    for (int c = lane; c < cols; c += 32) { float v = rp[c]; s += v * v; }
    for (int o = 16; o; o >>= 1) s += __shfl_xor(s, o, 32);
    float inv = 1.0f / fmaxf(sqrtf(s), 1e-12f);
    float* op = out + (long)row * cols;
    for (int c = lane; c < cols; c += 32) op[c] = rp[c] * inv;
}

// ------------------------------------------------------- encoder primitives

__global__ void k_tanh_copy(const float* __restrict__ src, int ss, float* __restrict__ dst,
                            int ds, long total, int cols) {
    long i = (long)blockIdx.x * blockDim.x + threadIdx.x;
    if (i >= total) return;
    long r = i / cols; int c = (int)(i % cols);
    dst[r * ds + c] = tanhf(src[r * ss + c]);
}

__global__ void k_edge_logits(const float* __restrict__ rels, const float* __restrict__ attn,
                              float* __restrict__ logits, int nedges) {
    int e = blockIdx.x * blockDim.x + threadIdx.x;
    if (e >= nedges) return;
    const float4* rp = (const float4*)(rels + (long)e * Dd);
    const float4* ap = (const float4*)attn;
    float s = 0.0f;
    for (int q = 0; q < 25; q++) {
        float4 r = rp[q], a = ap[q];
        s += r.x * a.x + r.y * a.y + r.z * a.z + r.w * a.w;
    }
    logits[e] = s;
}

__global__ void k_segmax(const int* __restrict__ adj, const float* __restrict__ logits,
                         float* __restrict__ segm, int nedges) {
    int e = blockIdx.x * blockDim.x + threadIdx.x;
    if (e < nedges) atomicMaxF(&segm[adj[2 * e]], logits[e]);
}

__global__ void k_segexp(const int* __restrict__ adj, float* __restrict__ logits,
                         const float* __restrict__ segm, float* __restrict__ segs, int nedges) {
    int e = blockIdx.x * blockDim.x + threadIdx.x;
    if (e >= nedges) return;
    int row = adj[2 * e];
    float w = expf(logits[e] - segm[row]);
    logits[e] = w;
    atomicAdd(&segs[row], w);
}

// wave-per-edge: Householder reflect neighbor feature by rels row, scale by
// softmax att, scatter-add into acc[row]. Dot via __shfl_xor wave reduction.
__global__ void k_aggregate(const int* __restrict__ adj, const float* __restrict__ rels,
                            const float* __restrict__ feats, int fstride,
                            const float* __restrict__ wexp, const float* __restrict__ segs,
                            float* __restrict__ acc, int nedges) {
    int wid = threadIdx.x >> 5, lane = threadIdx.x & 31;
    int e = blockIdx.x * (blockDim.x >> 5) + wid;
    if (e >= nedges) return;
    int row = adj[2 * e], col = adj[2 * e + 1];
    float att = wexp[e] / segs[row];
    const float* rr = rels + (long)e * Dd;
    const float* fr = feats + (long)col * fstride;
    float pd = 0.0f;
    for (int d = lane; d < Dd; d += 32) pd += fr[d] * rr[d];
    for (int o = 16; o; o >>= 1) pd += __shfl_xor(pd, o, 32);
    float c2 = 2.0f * pd;
    for (int d = lane; d < Dd; d += 32) {
        float v = fr[d] - c2 * rr[d];
        atomicAdd(&acc[(long)row * Dd + d], att * v);
    }
}

// ------------------------------------- proxy attention: pa/pf per out row
// block = 64 threads (2 waves) per node row; LDS caches the 300-wide row.
__global__ void k_proxy(const float* __restrict__ outb, const float* __restrict__ proxyn,
                        const float* __restrict__ proxy, float* __restrict__ pf, int rows) {
    __shared__ float srow[DOUTd];
    __shared__ float sred[64];
    __shared__ float spa[64];
    int row = blockIdx.x, tid = threadIdx.x;
    if (row >= rows) return;
    const float* orow = outb + (long)row * DOUTd;
    for (int c = tid; c < DOUTd; c += 64) srow[c] = orow[c];
    __syncthreads();
    // ||out_row||
    float ps = 0.0f;
    for (int c = tid; c < DOUTd; c += 64) { float v = srow[c]; ps += v * v; }
    sred[tid] = ps; __syncthreads();
    if (tid < 32) sred[tid] += sred[tid + 32];
    __syncthreads();
    if (tid < 32) {
        float v = sred[tid];
        for (int o = 16; o; o >>= 1) v += __shfl_xor(v, o, 32);
        if (tid == 0) sred[0] = v;
    }
    __syncthreads();
    float invn = 1.0f / fmaxf(sqrtf(sred[0]), 1e-12f);
    __syncthreads();
    // logits vs 64 normalized proxies, thread j owns proxy row j
    const float* pr = proxyn + tid * DOUTd;
    float lg = 0.0f;
    for (int c = 0; c < DOUTd; c++) lg += srow[c] * pr[c];
    lg *= invn;
    // softmax over 64
    sred[tid] = lg; __syncthreads();
    if (tid < 32) sred[tid] = fmaxf(sred[tid], sred[tid + 32]);
    __syncthreads();
    if (tid < 32) {
        float v = sred[tid];
        for (int o = 16; o; o >>= 1) v = fmaxf(v, __shfl_xor(v, o, 32));
        if (tid == 0) sred[0] = v;
    }
    __syncthreads();
    float m = sred[0];
    float e = expf(lg - m);
    __syncthreads();
    sred[tid] = e; __syncthreads();
    if (tid < 32) sred[tid] += sred[tid + 32];
    __syncthreads();
    if (tid < 32) {
        float v = sred[tid];
        for (int o = 16; o; o >>= 1) v += __shfl_xor(v, o, 32);
        if (tid == 0) sred[0] = v;
    }
    __syncthreads();
    spa[tid] = e / sred[0];
    __syncthreads();
    // pf = out - pa @ proxy
    float* prow = pf + (long)row * DOUTd;
    for (int c = tid; c < DOUTd; c += 64) {
        float a = 0.0f;
        for (int j = 0; j < NPROXY; j++) a += spa[j] * proxy[j * DOUTd + c];
        prow[c] = srow[c] - a;
    }
}

// ---------------- gate_k packing: pad to NPAD cols and interleave K-pairs.
// gkpad[k2][n][p] = gate_k[2*k2 + p][n]  (n >= 300 -> 0), k2 in [0,150)
// This makes each lane's B fragment (B[k][n], B[k+1][n]) one aligned 8B load.
__global__ void k_pack_gate(const float* __restrict__ gk, float* __restrict__ gkpad) {
    int t = blockIdx.x * blockDim.x + threadIdx.x;     // t = k2 * NPAD + n
    if (t >= (DOUTd / 2) * NPAD) return;
    int k2 = t / NPAD, n = t % NPAD;
    float a = (n < DOUTd) ? gk[(2 * k2 + 0) * DOUTd + n] : 0.0f;
    float b = (n < DOUTd) ? gk[(2 * k2 + 1) * DOUTd + n] : 0.0f;
    gkpad[(long)k2 * (2 * NPAD) + 2 * n + 0] = a;
    gkpad[(long)k2 * (2 * NPAD) + 2 * n + 1] = b;
}

// --------------------------- WMMA f32 GEMM (pf @ gate_k) + fused gate epilogue
// Each wave computes a 16(M) x 64(N) tile, K looped in steps of 4 via
// V_WMMA_F32_16X16X4_F32. A frag per 16x4 F32 layout (lane half -> K pair),
// C frag per 16x16 f32 layout (VGPR i: M=i | M=i+8 across lane halves).
// B fragments come from the packed/interleaved gkpad: fully unconditional
// vector loads, no exec-mask branching in the K loop.
__global__ void k_gemm_gate(const float* __restrict__ pf, const float* __restrict__ gkpad,
                            const float* __restrict__ bias, const float* __restrict__ outb,
                            float* __restrict__ dst, int dstOff, float scale, int accum,
                            int mtiles) {
    int wid = threadIdx.x >> 5, lane = threadIdx.x & 31;
    int gw = blockIdx.x * (blockDim.x >> 5) + wid;
    if (gw >= mtiles * 5) return;
    int mt = gw / 5, nt = gw % 5;
    int m0 = mt * 16, n0 = nt * 64;
    int half = lane >> 4, lm = lane & 15;

    v8f zero = {0.f, 0.f, 0.f, 0.f, 0.f, 0.f, 0.f, 0.f};
    v8f acc[4] = {zero, zero, zero, zero};

    const float* arow = pf + (long)(m0 + lm) * DOUTd + 2 * half;
    // lane's B pointer: K-pair row (k0/2 + half), column n -> 2 contiguous floats
    const float* bbase = gkpad + (long)half * (2 * NPAD) + 2 * (n0 + lm);

    for (int k0 = 0; k0 < DOUTd; k0 += 4) {
        v2f a = *(const v2f*)(arow + k0);
        const float* bk = bbase + (long)(k0 / 2) * (2 * NPAD);
        v2f b0 = *(const v2f*)(bk + 2 * 0);
        v2f b1 = *(const v2f*)(bk + 2 * 16);
        v2f b2 = *(const v2f*)(bk + 2 * 32);
        v2f b3 = *(const v2f*)(bk + 2 * 48);
        acc[0] = __builtin_amdgcn_wmma_f32_16x16x4_f32(false, a, false, b0, (short)0, acc[0], false, false);
        acc[1] = __builtin_amdgcn_wmma_f32_16x16x4_f32(false, a, false, b1, (short)0, acc[1], false, false);
        acc[2] = __builtin_amdgcn_wmma_f32_16x16x4_f32(false, a, false, b2, (short)0, acc[2], false, false);
        acc[3] = __builtin_amdgcn_wmma_f32_16x16x4_f32(false, a, false, b3, (short)0, acc[3], false, false);
    }

#pragma unroll
    for (int j = 0; j < 4; j++) {
        int col = n0 + j * 16 + lm;
        if (col >= DOUTd) continue;
        float bc = bias[col];
#pragma unroll
        for (int i = 0; i < 8; i++) {
            int row = m0 + i + 8 * half;
            float x = acc[j][i] + bc;
            float g = 1.0f / (1.0f + expf(-x));
            long base = (long)row * DOUTd + col;
            float o = outb[base], p = pf[base];
            float r = g * o + (1.0f - g) * p;
            long di = (long)row * (2 * DOUTd) + dstOff + col;
            float w = scale * r;
            if (accum) dst[di] += w; else dst[di] = w;
        }
    }
}

// ------------------------------------------------------------- host helpers

static void enqueue_srsm(const int* idx, const float* emb, float* cnt, float* featbuf,
                         hipStream_t stream) {
    hipMemsetAsync(cnt, 0, Nn * sizeof(float), stream);
    k_count<<<cdiv(ADJn, 256), 256, 0, stream>>>(idx, cnt, ADJn);
    hipMemsetAsync(featbuf, 0, (size_t)Nn * Dd * sizeof(float), stream);
    k_srsm_scatter<<<cdiv((long)ADJn * 25, 256), 256, 0, stream>>>(idx, emb, cnt, featbuf, ADJn);
}

static void enqueue_rels(const int* sp_idx, const float* sp_val, const float* emb,
                         float* rels, hipStream_t stream) {
    hipMemsetAsync(rels, 0, (size_t)Ee * Dd * sizeof(float), stream);
    k_rels_scatter<<<cdiv((long)NNZn * 25, 256), 256, 0, stream>>>(sp_idx, sp_val, emb, rels, NNZn);
    k_l2norm_rows<<<cdiv(Ee, 8), 256, 0, stream>>>(rels, rels, Ee, Dd);
}

static void enqueue_encoder(const int* adj, const float* rels, const float* attn,
                            const float* gate, const float* proxy, const float* bias,
                            float* featbuf, float* outbuf, float* pf, float* logits,
                            float* segm, float* segs, float* proxyn, float* gkpad,
                            float* dst, int dstOff, float scale, int accum,
                            hipStream_t stream) {
    long nd = (long)Nn * Dd;
    // outs[0] = tanh(features) into cols [0,100)
    k_tanh_copy<<<cdiv(nd, 256), 256, 0, stream>>>(featbuf, Dd, outbuf, DOUTd, nd, Dd);
    for (int l = 0; l < 2; l++) {
        k_edge_logits<<<cdiv(Ee, 256), 256, 0, stream>>>(rels, attn + l * Dd, logits, Ee);
        k_fill<<<cdiv(Nn, 256), 256, 0, stream>>>(segm, -3.402823466e38f, Nn);
        hipMemsetAsync(segs, 0, Nn * sizeof(float), stream);
        k_segmax<<<cdiv(Ee, 256), 256, 0, stream>>>(adj, logits, segm, Ee);
        k_segexp<<<cdiv(Ee, 256), 256, 0, stream>>>(adj, logits, segm, segs, Ee);
        hipMemsetAsync(featbuf, 0, nd * sizeof(float), stream);  // featbuf reused as accumulator
        k_aggregate<<<cdiv(Ee, 8), 256, 0, stream>>>(adj, rels, outbuf + l * Dd, DOUTd,
                                                     logits, segs, featbuf, Ee);
        k_tanh_copy<<<cdiv(nd, 256), 256, 0, stream>>>(featbuf, Dd, outbuf + (l + 1) * Dd,
                                                       DOUTd, nd, Dd);
    }
    k_l2norm_rows<<<cdiv(NPROXY, 8), 256, 0, stream>>>(proxy, proxyn, NPROXY, DOUTd);
    k_proxy<<<Nn, 64, 0, stream>>>(outbuf, proxyn, proxy, pf, Nn);
    k_pack_gate<<<cdiv((DOUTd / 2) * NPAD, 256), 256, 0, stream>>>(gate, gkpad);
    k_gemm_gate<<<cdiv((long)(Nn / 16) * 5, 8), 256, 0, stream>>>(pf, gkpad, bias, outbuf, dst,
                                                                  dstOff, scale, accum, Nn / 16);
}

extern "C" void kernel_launch(void* const* d_in, const int* in_sizes, int n_in,
                              void* d_out, int out_size, void* d_ws, size_t ws_size,
                              hipStream_t stream) {
    (void)in_sizes; (void)n_in; (void)out_size; (void)ws_size;
    const int*   adj_input = (const int*)d_in[0];
    const int*   r_index   = (const int*)d_in[1];
    const float* r_val     = (const float*)d_in[2];
    const int*   t_index   = (const int*)d_in[3];
    const int*   ent_adj   = (const int*)d_in[4];
    const int*   rel_adj   = (const int*)d_in[5];
    const int*   time_adj  = (const int*)d_in[6];
    const float* ent_emb   = (const float*)d_in[7];
    const float* rel_emb   = (const float*)d_in[8];
    const float* time_emb  = (const float*)d_in[9];
    const float* e_attn    = (const float*)d_in[10];
    const float* e_gate    = (const float*)d_in[11];
    const float* e_proxy   = (const float*)d_in[12];
    const float* e_bias    = (const float*)d_in[13];
    const float* r_attn    = (const float*)d_in[14];
    const float* r_gate    = (const float*)d_in[15];
    const float* r_proxy   = (const float*)d_in[16];
    const float* r_bias    = (const float*)d_in[17];
    float* out = (float*)d_out;

    // workspace carve-up (floats): ~444 MB total
    float* ws      = (float*)d_ws;
    float* rels    = ws;                  // E * D        = 40,000,000
    float* outbuf  = rels + 40000000;     // N * DOUT     = 30,000,000
    float* pf      = outbuf + 30000000;   // N * DOUT     = 30,000,000
    float* featbuf = pf + 30000000;       // N * D        = 10,000,000
    float* cnt     = featbuf + 10000000;  // N
    float* logits  = cnt + Nn;            // E
    float* segm    = logits + Ee;         // N
    float* segs    = segm + Nn;           // N
    float* proxyn  = segs + Nn;           // 64 * DOUT    = 19,200
    float* gkpad   = proxyn + NPROXY * DOUTd;  // 150 * 640 = 96,000

    // --- encoder ent: rels from (r_index, r_val, rel_emb); out cols [0,300)
    enqueue_rels(r_index, r_val, rel_emb, rels, stream);
    enqueue_srsm(ent_adj, ent_emb, cnt, featbuf, stream);
    enqueue_encoder(adj_input, rels, e_attn, e_gate, e_proxy, e_bias,
                    featbuf, outbuf, pf, logits, segm, segs, proxyn, gkpad,
                    out, 0, 1.0f, 0, stream);

    // --- encoder rel: same rels; writes 0.5*res into cols [300,600)
    enqueue_srsm(rel_adj, rel_emb, cnt, featbuf, stream);
    enqueue_encoder(adj_input, rels, r_attn, r_gate, r_proxy, r_bias,
                    featbuf, outbuf, pf, logits, segm, segs, proxyn, gkpad,
                    out, DOUTd, 0.5f, 0, stream);

    // --- encoder time: rels from (t_index, r_val, time_emb); adds 0.5*res
    enqueue_rels(t_index, r_val, time_emb, rels, stream);
    enqueue_srsm(time_adj, time_emb, cnt, featbuf, stream);
    enqueue_encoder(adj_input, rels, e_attn, e_gate, e_proxy, e_bias,
                    featbuf, outbuf, pf, logits, segm, segs, proxyn, gkpad,
                    out, DOUTd, 0.5f, 1, stream);
}